// Pct_70643622085271
// MI455X (gfx1250) — compile-verified
//
#include <hip/hip_runtime.h>
#include <hip/hip_bf16.h>

typedef __attribute__((ext_vector_type(2))) float v2f;
typedef __attribute__((ext_vector_type(8))) float v8f;
typedef __attribute__((ext_vector_type(4))) unsigned int v4u;
typedef __attribute__((ext_vector_type(8))) int v8i;
typedef __attribute__((ext_vector_type(4))) int v4i;

#define C64 64
#define KNN_K 16
#define LDSP 68  // padded LDS row stride (floats): 64 data + 4 pad

#if defined(__HIP_DEVICE_COMPILE__) && __has_builtin(__builtin_amdgcn_tensor_load_to_lds)
#define HAVE_TDM 1
#else
#define HAVE_TDM 0
#endif

// ---------------------------------------------------------------------------
// Exact KNN: one query point per thread, candidate tiles staged via LDS.
// Distance uses the exact reference formula qsq - 2*dot + csq so the self
// distance is exactly 0.0 and ordering matches top_k(-d, 16) (stable ties).
// ---------------------------------------------------------------------------
__global__ void knn_kernel(const float* __restrict__ coords, int N,
                           int* __restrict__ idxOut) {
  const int q = blockIdx.x * 256 + threadIdx.x;
  float qx = 0.f, qy = 0.f, qz = 0.f;
  if (q < N) {
    qx = coords[(size_t)q * 3 + 0];
    qy = coords[(size_t)q * 3 + 1];
    qz = coords[(size_t)q * 3 + 2];
  }
  const float qsq = qx * qx + qy * qy + qz * qz;

  float bd[KNN_K];
  int bi[KNN_K];
#pragma unroll
  for (int t = 0; t < KNN_K; ++t) { bd[t] = 3.4e38f; bi[t] = 0; }

  __shared__ float4 sc[256];

  for (int base = 0; base < N; base += 256) {
    const int c = base + (int)threadIdx.x;
    float4 p;
    if (c < N) {
      const float x = coords[(size_t)c * 3 + 0];
      const float y = coords[(size_t)c * 3 + 1];
      const float z = coords[(size_t)c * 3 + 2];
      p = make_float4(x, y, z, x * x + y * y + z * z);
    } else {
      p = make_float4(0.f, 0.f, 0.f, 3.4e38f);
    }
    __syncthreads();
    sc[threadIdx.x] = p;
    __syncthreads();

    if (base + 256 < N)  // speculative prefetch of next tile (global_prefetch_b8)
      __builtin_prefetch(coords + (size_t)(base + 256 + threadIdx.x) * 3, 0, 0);

    const int lim = (N - base < 256) ? (N - base) : 256;
    for (int j = 0; j < lim; ++j) {
      const float4 t = sc[j];
      const float d = qsq - 2.f * (qx * t.x + qy * t.y + qz * t.z) + t.w;
      if (d < bd[KNN_K - 1]) {
        bd[KNN_K - 1] = d;
        bi[KNN_K - 1] = base + j;
#pragma unroll
        for (int s = KNN_K - 1; s > 0; --s) {
          if (bd[s] < bd[s - 1]) {
            const float td = bd[s]; bd[s] = bd[s - 1]; bd[s - 1] = td;
            const int ti = bi[s];   bi[s] = bi[s - 1]; bi[s - 1] = ti;
          }
        }
      }
    }
  }

  if (q < N) {
#pragma unroll
    for (int t = 0; t < KNN_K; ++t) idxOut[(size_t)q * KNN_K + t] = bi[t];
  }
}

#if HAVE_TDM
// ---------------------------------------------------------------------------
// Issue one TDM 2D tile load (Global -> LDS) per calling wave.
// D# built per CDNA5 ISA:
//   group0: count=1 | lds_addr | global_addr(57b) | type=2
//   group1: d0 = data_size=2 (4B) | pad_enable | pad_interval=5 (64 DW)
//                | pad_amount=3 (4 DW)  => LDS rows padded 64 -> 68 floats
//           tensor_dim0/1, tile_dim0/1, tensor_dim0_stride
// Rows past tensor_dim1 are OOB -> LDS filled with zero (handles N edge).
// ---------------------------------------------------------------------------
#define TDM_G1D0_PAD68 0x07520000u  // data_size=2, pad_en, interval=5, amount=3

__device__ __forceinline__ void tdm_load_2d(unsigned ldsAddr, const void* gptr,
                                            unsigned dim1, unsigned g1d0) {
  const unsigned long long ga = (unsigned long long)gptr;
  v4u g0;
  g0.x = 1u;                                   // count=1, user-mode descriptor
  g0.y = ldsAddr;                              // LDS byte address
  g0.z = (unsigned)ga;                         // global_addr[31:0]
  g0.w = (unsigned)((ga >> 32) & 0x01FFFFFFull) | (2u << 30);  // [56:32]|type=2
  v8i g1;
  g1[0] = (int)g1d0;                           // flags / data_size / padding
  g1[1] = (int)(64u << 16);                    // tensor_dim0 = 64 (low16)
  g1[2] = (int)((dim1 & 0xFFFFu) << 16);       // dim0 hi=0 | tensor_dim1 lo16
  g1[3] = (int)((dim1 >> 16) | (64u << 16));   // tensor_dim1 hi16 | tile_dim0=64
  g1[4] = 64;                                  // tile_dim1 = 64 rows
  g1[5] = 64;                                  // tensor_dim0_stride = 64 elems
  g1[6] = 0;
  g1[7] = 0;
  const v4i gz = {0, 0, 0, 0};
#if defined(__clang_major__) && __clang_major__ >= 23
  const v8i gz8 = {0, 0, 0, 0, 0, 0, 0, 0};
  __builtin_amdgcn_tensor_load_to_lds(g0, g1, gz, gz, gz8, 0);
#else
  __builtin_amdgcn_tensor_load_to_lds(g0, g1, gz, gz, 0);
#endif
}
#endif  // HAVE_TDM

// ---------------------------------------------------------------------------
// Out[N,64] = X[N,64] @ W[64,64]^T + bias[64]   (C[n][o] = sum_c X[n][c]W[o][c])
// 128 threads = 4 waves; block owns a 64-row stripe; wave wv owns cols
// [16*wv, 16*wv+16). X and W tiles staged to LDS by the Tensor Data Mover
// (padded to 68-float rows in hardware), then each wave hoists its 16 B
// fragments (W) into registers once and reuses them across 4 M-tiles:
// 64 x V_WMMA_F32_16X16X4_F32 per wave.
// Fragment mapping per ISA layouts: A: lane&15 = M, (lane>>4)*2 = K pair;
// B: lane&15 = N, same K split; D: VGPR r -> row r + 8*(lane>>4), lane&15 = col.
// ---------------------------------------------------------------------------
__global__ void gemm64_kernel(const float* __restrict__ X,
                              const float* __restrict__ W,
                              const float* __restrict__ bias,
                              float* __restrict__ Out, int N) {
  __shared__ float lX[64 * LDSP];
  __shared__ float lW[64 * LDSP];

  const int tid = threadIdx.x;  // 128
  const int n0 = blockIdx.x * 64;

#if HAVE_TDM
  {
    const int wv0 = tid >> 5;
    if (wv0 == 0) {
      tdm_load_2d((unsigned)(unsigned long long)(void*)lX,
                  X + (size_t)n0 * C64, (unsigned)(N - n0), TDM_G1D0_PAD68);
      __builtin_amdgcn_s_wait_tensorcnt(0);
    } else if (wv0 == 1) {
      tdm_load_2d((unsigned)(unsigned long long)(void*)lW,
                  W, 64u, TDM_G1D0_PAD68);
      __builtin_amdgcn_s_wait_tensorcnt(0);
    }
  }
#else
  for (int i = tid; i < 64 * 64; i += 128) {
    const int r = i >> 6, c = i & 63;
    const int row = n0 + r;
    lX[r * LDSP + c] = (row < N) ? X[(size_t)row * C64 + c] : 0.f;
    lW[r * LDSP + c] = W[i];
  }
#endif
  __syncthreads();

  const int wv = tid >> 5;
  const int lane = tid & 31;
  const int o0 = wv * 16;
  const int mn = lane & 15;       // M for A, N for B/D
  const int kb = (lane >> 4) * 2; // K sub-pair select

  // Hoist this wave's B fragments (full K=64) into registers: 16 x v2f.
  v2f bf[16];
#pragma unroll
  for (int kk = 0; kk < 16; ++kk) {
    bf[kk].x = lW[(o0 + mn) * LDSP + kk * 4 + kb];
    bf[kk].y = lW[(o0 + mn) * LDSP + kk * 4 + kb + 1];
  }

  const int col = o0 + mn;
  const float bv = bias[col];

#pragma unroll
  for (int mt = 0; mt < 4; ++mt) {
    const int mrow = mt * 16 + mn;
    v8f acc = {0.f, 0.f, 0.f, 0.f, 0.f, 0.f, 0.f, 0.f};
#pragma unroll
    for (int kk = 0; kk < 16; ++kk) {
      v2f a;
      a.x = lX[mrow * LDSP + kk * 4 + kb];
      a.y = lX[mrow * LDSP + kk * 4 + kb + 1];
      acc = __builtin_amdgcn_wmma_f32_16x16x4_f32(
          /*neg_a=*/false, a, /*neg_b=*/false, bf[kk],
          /*c_mod=*/(short)0, acc, /*reuse_a=*/false, /*reuse_b=*/false);
    }
    const int rbase = n0 + mt * 16 + ((lane >> 4) * 8);
    if (rbase + 7 < N) {
#pragma unroll
      for (int r = 0; r < 8; ++r)
        Out[(size_t)(rbase + r) * C64 + col] = acc[r] + bv;
    } else {
#pragma unroll
      for (int r = 0; r < 8; ++r)
        if (rbase + r < N) Out[(size_t)(rbase + r) * C64 + col] = acc[r] + bv;
    }
  }
}

// ---------------------------------------------------------------------------
// Attention: one wave (32 lanes) per point; lane owns channel pair {2L,2L+1}.
// att[k] = dot64(Kf[idx[n,k]], Qf[n]) * (1/8); softmax; y = x - sum att[k]*Vf.
// ---------------------------------------------------------------------------
__global__ void attn_kernel(const float* __restrict__ X,
                            const float* __restrict__ Qf,
                            const float* __restrict__ Kf,
                            const float* __restrict__ Vf,
                            const int* __restrict__ idx,
                            float* __restrict__ Y, int N) {
  const int n = blockIdx.x * 8 + (threadIdx.x >> 5);
  const int lane = threadIdx.x & 31;
  if (n >= N) return;

  const float2 qv = *(const float2*)(Qf + (size_t)n * C64 + lane * 2);
  const int nb = idx[(size_t)n * KNN_K + (lane & 15)];

  float att[KNN_K];
#pragma unroll
  for (int k = 0; k < KNN_K; ++k) {
    const int j = __shfl(nb, k, 32);
    const float2 kf = *(const float2*)(Kf + (size_t)j * C64 + lane * 2);
    float p = qv.x * kf.x + qv.y * kf.y;
    p += __shfl_xor(p, 16, 32);
    p += __shfl_xor(p, 8, 32);
    p += __shfl_xor(p, 4, 32);
    p += __shfl_xor(p, 2, 32);
    p += __shfl_xor(p, 1, 32);
    att[k] = p * 0.125f;  // 1/sqrt(64)
  }

  float m = att[0];
#pragma unroll
  for (int k = 1; k < KNN_K; ++k) m = fmaxf(m, att[k]);
  float s = 0.f;
#pragma unroll
  for (int k = 0; k < KNN_K; ++k) { att[k] = __expf(att[k] - m); s += att[k]; }
  const float inv = 1.f / s;

  float ax = 0.f, ay = 0.f;
#pragma unroll
  for (int k = 0; k < KNN_K; ++k) {
    const int j = __shfl(nb, k, 32);
    const float2 vf = *(const float2*)(Vf + (size_t)j * C64 + lane * 2);
    const float w = att[k] * inv;
    ax += w * vf.x;
    ay += w * vf.y;
  }

  const float2 xv = *(const float2*)(X + (size_t)n * C64 + lane * 2);
  float2 out;
  out.x = xv.x - ax;
  out.y = xv.y - ay;
  *(float2*)(Y + (size_t)n * C64 + lane * 2) = out;
}

// ---------------------------------------------------------------------------
// Deterministic BN reduction, pass 1: per-block per-channel sum & sumsq.
// part[0 .. R*64) = sums, part[R*64 .. 2*R*64) = sums of squares.
// ---------------------------------------------------------------------------
__global__ void bn_partial_kernel(const float* __restrict__ Y2,
                                  float* __restrict__ part, int N, int R) {
  const int c = threadIdx.x & 63;
  const int wkr = threadIdx.x >> 6;  // 0..3
  float s = 0.f, q = 0.f;
  for (int r = blockIdx.x * 4 + wkr; r < N; r += R * 4) {
    const float v = Y2[(size_t)r * C64 + c];
    s += v;
    q += v * v;
  }
  __shared__ float ls[256], lq[256];
  ls[threadIdx.x] = s;
  lq[threadIdx.x] = q;
  __syncthreads();
  if (threadIdx.x < 64) {
    s = ls[c] + ls[c + 64] + ls[c + 128] + ls[c + 192];
    q = lq[c] + lq[c + 64] + lq[c + 128] + lq[c + 192];
    part[(size_t)blockIdx.x * 64 + c] = s;
    part[(size_t)R * 64 + (size_t)blockIdx.x * 64 + c] = q;
  }
}

// Pass 2: stats[c]=mu, stats[64+c]=rsqrt(var+eps). Fixed-order sum.
__global__ void bn_final_kernel(const float* __restrict__ part,
                                float* __restrict__ stats, int N, int R) {
  const int c = threadIdx.x;  // 64 threads
  float s = 0.f, q = 0.f;
  for (int r = 0; r < R; ++r) {
    s += part[(size_t)r * 64 + c];
    q += part[(size_t)R * 64 + (size_t)r * 64 + c];
  }
  const float invN = 1.f / (float)N;
  const float mu = s * invN;
  const float var = q * invN - mu * mu;
  stats[c] = mu;
  stats[64 + c] = rsqrtf(var + 1e-5f);
}

// out = X + relu((Y2 - mu)*inv*g + b)
__global__ void bn_apply_kernel(const float* __restrict__ X,
                                const float* __restrict__ Y2,
                                const float* __restrict__ stats,
                                const float* __restrict__ g,
                                const float* __restrict__ b,
                                float* __restrict__ Out, int total) {
  const int i = blockIdx.x * 256 + threadIdx.x;
  if (i >= total) return;
  const int c = i & 63;
  float v = (Y2[i] - stats[c]) * stats[64 + c] * g[c] + b[c];
  v = fmaxf(v, 0.f);
  Out[i] = X[i] + v;
}

// ---------------------------------------------------------------------------
extern "C" void kernel_launch(void* const* d_in, const int* in_sizes, int n_in,
                              void* d_out, int out_size, void* d_ws, size_t ws_size,
                              hipStream_t stream) {
  const float* coords = (const float*)d_in[0];  // [N,3]
  const float* feats  = (const float*)d_in[1];  // [N,64]
  const float* qkv_w  = (const float*)d_in[2];  // [2,3,64,64]
  const float* qkv_b  = (const float*)d_in[3];  // [2,3,64]
  const float* conv_w = (const float*)d_in[4];  // [2,64,64]
  const float* conv_b = (const float*)d_in[5];  // [2,64]
  const float* bn_g   = (const float*)d_in[6];  // [2,64]
  const float* bn_b   = (const float*)d_in[7];  // [2,64]
  float* out = (float*)d_out;

  const int N = in_sizes[1] / C64;
  const int R = 128;  // BN partial-reduction blocks

  // Workspace layout (16B aligned chunks)
  char* w = (char*)d_ws;
  int* idx = (int*)w;            w += (size_t)N * KNN_K * sizeof(int);
  float* bufA = (float*)w;       w += (size_t)N * C64 * sizeof(float);
  float* bufB = (float*)w;       w += (size_t)N * C64 * sizeof(float);
  float* bufC = (float*)w;       w += (size_t)N * C64 * sizeof(float);
  float* bufD = (float*)w;       w += (size_t)N * C64 * sizeof(float);
  float* bufE = (float*)w;       w += (size_t)N * C64 * sizeof(float);
  float* part = (float*)w;       w += (size_t)R * 64 * 2 * sizeof(float);
  float* stats = (float*)w;      w += 128 * sizeof(float);

  const int gemmGrid = (N + 63) / 64;
  const int attnGrid = (N + 7) / 8;
  const int elemGrid = (N * C64 + 255) / 256;

  // KNN (coords fixed across layers -> single pass)
  knn_kernel<<<(N + 255) / 256, 256, 0, stream>>>(coords, N, idx);

  // ---- layer 0 : X = feats ----
  gemm64_kernel<<<gemmGrid, 128, 0, stream>>>(feats, qkv_w + 0 * 4096, qkv_b + 0 * 64, bufA, N); // Q
  gemm64_kernel<<<gemmGrid, 128, 0, stream>>>(feats, qkv_w + 1 * 4096, qkv_b + 1 * 64, bufB, N); // K
  gemm64_kernel<<<gemmGrid, 128, 0, stream>>>(feats, qkv_w + 2 * 4096, qkv_b + 2 * 64, bufC, N); // V
  attn_kernel<<<attnGrid, 256, 0, stream>>>(feats, bufA, bufB, bufC, idx, bufD, N);              // y
  gemm64_kernel<<<gemmGrid, 128, 0, stream>>>(bufD, conv_w + 0, conv_b + 0, bufE, N);            // y2
  bn_partial_kernel<<<R, 256, 0, stream>>>(bufE, part, N, R);
  bn_final_kernel<<<1, 64, 0, stream>>>(part, stats, N, R);
  bn_apply_kernel<<<elemGrid, 256, 0, stream>>>(feats, bufE, stats, bn_g + 0, bn_b + 0, bufA, N * C64);

  // ---- layer 1 : X = bufA ----
  gemm64_kernel<<<gemmGrid, 128, 0, stream>>>(bufA, qkv_w + 3 * 4096, qkv_b + 3 * 64, bufB, N);  // Q
  gemm64_kernel<<<gemmGrid, 128, 0, stream>>>(bufA, qkv_w + 4 * 4096, qkv_b + 4 * 64, bufC, N);  // K
  gemm64_kernel<<<gemmGrid, 128, 0, stream>>>(bufA, qkv_w + 5 * 4096, qkv_b + 5 * 64, bufD, N);  // V
  attn_kernel<<<attnGrid, 256, 0, stream>>>(bufA, bufB, bufC, bufD, idx, bufE, N);               // y
  gemm64_kernel<<<gemmGrid, 128, 0, stream>>>(bufE, conv_w + 4096, conv_b + 64, bufB, N);        // y2
  bn_partial_kernel<<<R, 256, 0, stream>>>(bufB, part, N, R);
  bn_final_kernel<<<1, 64, 0, stream>>>(part, stats, N, R);
  bn_apply_kernel<<<elemGrid, 256, 0, stream>>>(bufA, bufB, stats, bn_g + 64, bn_b + 64, out, N * C64);
}